// GIN_80453327388881
// MI455X (gfx1250) — compile-verified
//
#include <hip/hip_runtime.h>
#include <hip/hip_bf16.h>

// ---------------------------------------------------------------------------
// GIN forward for MI455X (gfx1250, wave32).
// GEMMs use V_WMMA_F32_16X16X4_F32 (fp32 WMMA) since layer outputs are sign-
// binarized and must stay numerically faithful to the fp32 reference.
// W panels are staged to LDS via gfx1250 async global->LDS copies.
// ---------------------------------------------------------------------------

typedef float v2f __attribute__((ext_vector_type(2)));
typedef float v8f __attribute__((ext_vector_type(8)));

#define N_NODES 65536
#define N_EDGES 1048576
#define N_FEAT  128
#define HIDDEN  256
#define N_GRAPHS 512
#define FC1_IN  (3 * 3 * HIDDEN)   // 2304
#define CAT_W   (3 * HIDDEN)       // 768

#define KT    64    // K panel staged in LDS per iteration
#define LDSW  72    // padded LDS row stride (floats): rows 2 apart -> +16 banks

#ifndef __has_builtin
#define __has_builtin(x) 0
#endif
#if __has_builtin(__builtin_amdgcn_global_load_async_to_lds_b128)
#define HAVE_ASYNC_LDS 1
#else
#define HAVE_ASYNC_LDS 0
#endif

#if HAVE_ASYNC_LDS
// Exact parameter types per hipcc diagnostic:
//   arg0: int __attribute__((vector_size(16))) addrspace(1)*  (non-const)
//   arg1: LDS-side pointer (AS3); typed AS3 v4i* also converts to AS3 void*
typedef int async_v4i __attribute__((vector_size(4 * sizeof(int))));
typedef __attribute__((address_space(1))) async_v4i* async_gptr;
typedef __attribute__((address_space(3))) async_v4i* async_lptr;
#if __has_builtin(__builtin_amdgcn_s_wait_asynccnt)
#define ASYNC_WAIT() __builtin_amdgcn_s_wait_asynccnt(0)
#else
#define ASYNC_WAIT() asm volatile("s_wait_asynccnt 0x0" ::: "memory")
#endif
#endif

static __device__ __forceinline__ v8f wmma4(v2f a, v2f b, v8f c) {
  // D = A(16x4 f32) * B(4x16 f32) + C(16x16 f32)
  return __builtin_amdgcn_wmma_f32_16x16x4_f32(
      /*neg_a=*/false, a, /*neg_b=*/false, b,
      /*c_mod=*/(short)0, c, /*reuse_a=*/false, /*reuse_b=*/false);
}

// ---------------------------------------------------------------------------
// Edge scatter-add: one wave (32 lanes) per edge, float4 gather + b32 atomics.
// aggr[dst, f] += h[src, f]  for f in [0, D)
// ---------------------------------------------------------------------------
__global__ __launch_bounds__(256)
void gin_aggr(const float* __restrict__ h, int ldh,
              const int* __restrict__ src, const int* __restrict__ dst,
              float* __restrict__ aggr, int D, int E) {
  int widx = (blockIdx.x * blockDim.x + threadIdx.x) >> 5;
  int lane = threadIdx.x & 31;
  if (widx >= E) return;
  int s = src[widx];
  int d = dst[widx];
  const float* hs = h + (size_t)s * ldh;
  float* ad = aggr + (size_t)d * D;
  for (int f = lane * 4; f < D; f += 128) {
    float4 v = *(const float4*)(hs + f);
    atomicAdd(&ad[f + 0], v.x);
    atomicAdd(&ad[f + 1], v.y);
    atomicAdd(&ad[f + 2], v.z);
    atomicAdd(&ad[f + 3], v.w);
  }
}

// ---------------------------------------------------------------------------
// aggr[i] = (1+eps) * x[i] + aggr[i]   (elementwise, aggr row stride == D)
// ---------------------------------------------------------------------------
__global__ __launch_bounds__(256)
void gin_combine(const float* __restrict__ x, int ldx,
                 float* __restrict__ aggr, int D, int n_total,
                 const float* __restrict__ eps) {
  int i = blockIdx.x * blockDim.x + threadIdx.x;
  if (i >= n_total) return;
  int row = i / D;
  int col = i - row * D;
  float e = 1.0f + eps[0];
  aggr[i] = fmaf(e, x[(size_t)row * ldx + col], aggr[i]);
}

// ---------------------------------------------------------------------------
// WMMA GEMM: out[m, ocol+n] = epi(A[m,:K] @ W[:K, n] + bias[n])
//   W is K x 256 row-major. Block = 8 waves; wave tile = 32(M) x 64(N);
//   block tile = 256(M) x 64(N). Grid = (256/64, M/256).
//   W panel [KT x 64] staged in LDS (padded stride LDSW, bank-conflict-free).
//   mode 0: ReLU (GEMM1). mode 1: binarize h>0 (GEMM2).
// Fragment layouts per CDNA5 ISA 7.12.2 (wave32):
//   A 16x4 : lanes 0-15 -> row M=lane, {K=k,k+1}; lanes 16-31 -> {K=k+2,k+3}
//   B 4x16 : lanes 0-15 -> col N=lane, rows {k,k+1}; lanes 16-31 rows {k+2,k+3}
//   C 16x16: VGPR v -> M = v (+8 for lanes 16-31), N = lane & 15
// ---------------------------------------------------------------------------
__device__ __forceinline__
void gin_epilogue(const v8f& acc, int t, int col0, int lh, int mrow,
                  const float* __restrict__ bias, float* __restrict__ out,
                  int ldo, int ocol, int mode) {
  int col = col0 + t * 16 + lh;
  float bb = bias[col];
#pragma unroll
  for (int v = 0; v < 8; ++v) {
    float val = acc[v] + bb;
    val = (mode == 0) ? fmaxf(val, 0.0f) : ((val > 0.0f) ? 1.0f : 0.0f);
    out[(size_t)(mrow + v) * ldo + ocol + col] = val;
  }
}

__global__ __launch_bounds__(256)
void gin_gemm_wmma(const float* __restrict__ A, int lda,
                   const float* __restrict__ W,
                   const float* __restrict__ bias,
                   float* __restrict__ out, int ldo, int ocol,
                   int K, int mode) {
  __shared__ float sW[KT * LDSW];   // 18 KB

  const int tid  = threadIdx.x;
  const int lane = tid & 31;
  const int wave = tid >> 5;
  const int row0 = blockIdx.y * 256 + wave * 32;
  const int col0 = blockIdx.x * 64;
  const int lh   = lane & 15;
  const int khi  = (lane >> 4) * 2;   // 0 for lanes 0-15, 2 for lanes 16-31

  v8f acc[8];
#pragma unroll
  for (int i = 0; i < 8; ++i) acc[i] = (v8f){};

  const float* arow0 = A + (size_t)(row0 + lh) * lda;
  const float* arow1 = A + (size_t)(row0 + 16 + lh) * lda;

  for (int k0 = 0; k0 < K; k0 += KT) {
    // ---- stage W[k0:k0+KT, col0:col0+64] into LDS, 16B chunks per lane ----
#pragma unroll
    for (int i = tid; i < KT * 16; i += 256) {   // 16 x 16B chunks per row
      int r = i >> 4;
      int c = i & 15;
      const float* gsrc = W + (size_t)(k0 + r) * HIDDEN + col0 + c * 4;
#if HAVE_ASYNC_LDS
      __builtin_amdgcn_global_load_async_to_lds_b128(
          (async_gptr)const_cast<float*>(gsrc),
          (async_lptr)&sW[r * LDSW + c * 4],
          0, 0);
#else
      *(float4*)&sW[r * LDSW + c * 4] = *(const float4*)gsrc;
#endif
    }
#if HAVE_ASYNC_LDS
    ASYNC_WAIT();
#endif
    __syncthreads();

    // ---- inner K loop: 2 A frags (global) x 4 B frags (LDS) -> 8 WMMA ----
#pragma unroll 4
    for (int kk = 0; kk < KT; kk += 4) {
      const int kg = k0 + kk + khi;
      v2f a0, a1;
      a0.x = arow0[kg]; a0.y = arow0[kg + 1];
      a1.x = arow1[kg]; a1.y = arow1[kg + 1];
      const float* bs = &sW[(kk + khi) * LDSW + lh];
      v2f b0, b1, b2, b3;
      b0.x = bs[0];  b0.y = bs[LDSW + 0];
      b1.x = bs[16]; b1.y = bs[LDSW + 16];
      b2.x = bs[32]; b2.y = bs[LDSW + 32];
      b3.x = bs[48]; b3.y = bs[LDSW + 48];
      acc[0] = wmma4(a0, b0, acc[0]);
      acc[1] = wmma4(a0, b1, acc[1]);
      acc[2] = wmma4(a0, b2, acc[2]);
      acc[3] = wmma4(a0, b3, acc[3]);
      acc[4] = wmma4(a1, b0, acc[4]);
      acc[5] = wmma4(a1, b1, acc[5]);
      acc[6] = wmma4(a1, b2, acc[6]);
      acc[7] = wmma4(a1, b3, acc[7]);
    }
    __syncthreads();
  }

  const int mrow0 = row0 + (lane >> 4) * 8;
#pragma unroll
  for (int mf = 0; mf < 2; ++mf) {
#pragma unroll
    for (int t = 0; t < 4; ++t) {
      gin_epilogue(acc[mf * 4 + t], t, col0, lh, mrow0 + mf * 16,
                   bias, out, ldo, ocol, mode);
    }
  }
}

// ---------------------------------------------------------------------------
// Graph boundaries: batch is sorted, starts[g] = lower_bound(batch, g).
// ---------------------------------------------------------------------------
__global__ void gin_starts(const int* __restrict__ batch, int* __restrict__ starts) {
  int g = blockIdx.x * blockDim.x + threadIdx.x;
  if (g > N_GRAPHS) return;
  int lo = 0, hi = N_NODES;
  while (lo < hi) {
    int mid = (lo + hi) >> 1;
    if (batch[mid] < g) lo = mid + 1; else hi = mid;
  }
  starts[g] = lo;
}

// ---------------------------------------------------------------------------
// Segment mean/max/sum over contiguous node ranges (atomic-free).
// cat entries are exactly {0,1} -> seg_max = (seg_sum > 0).
// gfeat[g] = [mean(768) | max(768) | sum(768)]
// ---------------------------------------------------------------------------
__global__ __launch_bounds__(256)
void gin_seg(const float* __restrict__ cat, const int* __restrict__ starts,
             float* __restrict__ gfeat) {
  int g = blockIdx.x;
  int s = starts[g], e = starts[g + 1];
  float cnt = (float)(e - s);
  float inv = 1.0f / fmaxf(cnt, 1.0f);
  for (int f = threadIdx.x; f < CAT_W; f += 256) {
    float sum = 0.0f;
    for (int n = s; n < e; ++n) sum += cat[(size_t)n * CAT_W + f];
    size_t base = (size_t)g * FC1_IN;
    gfeat[base + f]             = sum * inv;                  // mean
    gfeat[base + CAT_W + f]     = (sum > 0.5f) ? 1.0f : 0.0f; // max of {0,1}
    gfeat[base + 2 * CAT_W + f] = sum;                        // sum
  }
}

// ---------------------------------------------------------------------------
// fc1: gh[g, j] = relu(gfeat[g,:] @ fc1_w[:, j] + fc1_b[j]);  2304 -> 256
// ---------------------------------------------------------------------------
__global__ __launch_bounds__(256)
void gin_fc1(const float* __restrict__ gfeat, const float* __restrict__ w,
             const float* __restrict__ b, float* __restrict__ gh) {
  __shared__ float sg[FC1_IN];
  int g = blockIdx.x;
  for (int i = threadIdx.x; i < FC1_IN; i += 256)
    sg[i] = gfeat[(size_t)g * FC1_IN + i];
  __syncthreads();
  int j = threadIdx.x;
  float acc = b[j];
  for (int k = 0; k < FC1_IN; ++k)
    acc = fmaf(sg[k], w[(size_t)k * HIDDEN + j], acc);
  gh[(size_t)g * HIDDEN + j] = fmaxf(acc, 0.0f);
}

// ---------------------------------------------------------------------------
// fc2 + sigmoid: out[g, c], 256 -> 10
// ---------------------------------------------------------------------------
__global__ void gin_fc2(const float* __restrict__ gh, const float* __restrict__ w,
                        const float* __restrict__ b, float* __restrict__ out) {
  int g = blockIdx.x;
  int c = threadIdx.x;
  if (c >= 10) return;
  float acc = b[c];
  for (int k = 0; k < HIDDEN; ++k)
    acc = fmaf(gh[(size_t)g * HIDDEN + k], w[k * 10 + c], acc);
  out[g * 10 + c] = 1.0f / (1.0f + expf(-acc));
}

// ---------------------------------------------------------------------------
extern "C" void kernel_launch(void* const* d_in, const int* in_sizes, int n_in,
                              void* d_out, int out_size, void* d_ws, size_t ws_size,
                              hipStream_t stream) {
  const float* x     = (const float*)d_in[0];
  const int*   eidx  = (const int*)d_in[1];
  const int*   batch = (const int*)d_in[2];
  const float* w1[3]  = {(const float*)d_in[3],  (const float*)d_in[8],  (const float*)d_in[13]};
  const float* b1[3]  = {(const float*)d_in[4],  (const float*)d_in[9],  (const float*)d_in[14]};
  const float* w2[3]  = {(const float*)d_in[5],  (const float*)d_in[10], (const float*)d_in[15]};
  const float* b2[3]  = {(const float*)d_in[6],  (const float*)d_in[11], (const float*)d_in[16]};
  const float* eps[3] = {(const float*)d_in[7],  (const float*)d_in[12], (const float*)d_in[17]};
  const float* fc1w = (const float*)d_in[18];
  const float* fc1b = (const float*)d_in[19];
  const float* fc2w = (const float*)d_in[20];
  const float* fc2b = (const float*)d_in[21];

  // Workspace carve (~326 MB total)
  char* ws = (char*)d_ws;
  float* aggr  = (float*)ws; ws += (size_t)N_NODES * HIDDEN * 4;   // 64 MB
  float* h1    = (float*)ws; ws += (size_t)N_NODES * HIDDEN * 4;   // 64 MB
  float* cat   = (float*)ws; ws += (size_t)N_NODES * CAT_W * 4;    // 192 MB
  int*   starts= (int*)ws;   ws += 1024 * 4;                       // 513 used
  float* gfeat = (float*)ws; ws += (size_t)N_GRAPHS * FC1_IN * 4;  // 4.5 MB
  float* gh    = (float*)ws; ws += (size_t)N_GRAPHS * HIDDEN * 4;  // 0.5 MB

  const int* src = eidx;
  const int* dst = eidx + N_EDGES;
  const dim3 ggrid(HIDDEN / 64, N_NODES / 256);   // (4, 256)

  for (int l = 0; l < 3; ++l) {
    const int din = (l == 0) ? N_FEAT : HIDDEN;
    const float* hin = (l == 0) ? x : (cat + (size_t)(l - 1) * HIDDEN);
    const int ldh = (l == 0) ? N_FEAT : CAT_W;

    (void)hipMemsetAsync(aggr, 0, (size_t)N_NODES * din * 4, stream);

    // scatter-add over edges: one wave per edge (8 waves / block)
    gin_aggr<<<N_EDGES / 8, 256, 0, stream>>>(hin, ldh, src, dst, aggr, din, N_EDGES);

    // aggr = (1+eps)*hin + aggr
    const int tot = N_NODES * din;
    gin_combine<<<tot / 256, 256, 0, stream>>>(hin, ldh, aggr, din, tot, eps[l]);

    // h1 = relu(aggr @ w1 + b1)
    gin_gemm_wmma<<<ggrid, 256, 0, stream>>>(aggr, din, w1[l], b1[l],
                                             h1, HIDDEN, 0, din, /*mode=*/0);
    // cat[:, l*256 : (l+1)*256] = (h1 @ w2 + b2 > 0) ? 1 : 0
    gin_gemm_wmma<<<ggrid, 256, 0, stream>>>(h1, HIDDEN, w2[l], b2[l],
                                             cat, CAT_W, l * HIDDEN, HIDDEN, /*mode=*/1);
  }

  // Readout
  gin_starts<<<3, 256, 0, stream>>>(batch, starts);
  gin_seg<<<N_GRAPHS, 256, 0, stream>>>(cat, starts, gfeat);
  gin_fc1<<<N_GRAPHS, 256, 0, stream>>>(gfeat, fc1w, fc1b, gh);
  gin_fc2<<<N_GRAPHS, 32, 0, stream>>>(gh, fc2w, fc2b, (float*)d_out);
}